// DCTGlobalFeatureExtractor_63273458204770
// MI455X (gfx1250) — compile-verified
//
#include <hip/hip_runtime.h>
#include <math.h>

typedef __attribute__((ext_vector_type(2))) float v2f;
typedef __attribute__((ext_vector_type(8))) float v8f;

#define NDIM 512
#define NUM_BINS 256
#define HIDDEN 512
#define FEATURE_DIM 256
#define BC 96

#define TILE_M 128
#define TILE_N 64
#define KB 16

// ---------------------------------------------------------------------------
// K0: build orthonormal DCT-II matrix (f32, matching jnp op order) + zero ws
// ---------------------------------------------------------------------------
__global__ __launch_bounds__(256) void dct_init_kernel(float* __restrict__ D,
                                                       int* __restrict__ maxv,
                                                       int* __restrict__ hist) {
  int g = blockIdx.x * 256 + threadIdx.x;
  if (g < NDIM * NDIM) {
    int k = g / NDIM, i = g % NDIM;
    // jnp: pi * (2i+1) * k / (2n), all f32
    float a = 3.14159265358979323846f * (2.0f * (float)i + 1.0f);
    a = a * (float)k;
    a = a / (2.0f * (float)NDIM);
    float s = (k == 0) ? sqrtf(1.0f / (float)NDIM) : sqrtf(2.0f / (float)NDIM);
    D[g] = s * cosf(a);
  }
  if (g < BC) maxv[g] = 0;
  if (g < BC * NUM_BINS) hist[g] = 0;
}

// ---------------------------------------------------------------------------
// K1/K2: fp32 WMMA GEMM  C = A @ B  (optionally B transposed; optional
// |.| epilogue with per-image atomic max).  M = N = K = 512.
// Block: 256 threads = 8 waves; block tile 128x64; wave tile 32x32
// (2x2 of v_wmma_f32_16x16x4_f32 accumulators).
// ---------------------------------------------------------------------------
__global__ __launch_bounds__(256) void dct_gemm_kernel(
    const float* __restrict__ A, const float* __restrict__ B,
    float* __restrict__ C,
    long aStride, long bStride, long cStride,
    int transB, int absMax, int* __restrict__ maxv) {
  __shared__ float As[TILE_M][KB + 1];   // 128x17 f32
  __shared__ float Bs[KB][TILE_N + 1];   // 16x65 f32
  __shared__ float red[256];

  const int img = blockIdx.z;
  const int m0 = blockIdx.y * TILE_M;
  const int n0 = blockIdx.x * TILE_N;
  const float* Ai = A + (size_t)aStride * img;
  const float* Bi = B + (size_t)bStride * img;
  float* Ci = C + (size_t)cStride * img;

  const int tid = threadIdx.x;
  const int lane = tid & 31;
  const int wave = tid >> 5;
  const int wm = wave & 3;   // wave M group: 0..3
  const int wn = wave >> 2;  // wave N group: 0..1
  const int lr = lane & 15;  // lane row/col within half-wave
  const int hi = lane >> 4;  // half-wave select

  v8f acc[2][2] = {};

  // cooperative LDS load indices
  const int arow = tid >> 1;        // 0..127
  const int acol = (tid & 1) * 8;   // 0 / 8
  const int brow = tid >> 4;        // 0..15
  const int bcol = (tid & 15) * 4;  // 0..60
  const int tcol = tid >> 2;        // 0..63  (transB path)
  const int trow = (tid & 3) * 4;   // 0,4,8,12

  for (int k0 = 0; k0 < NDIM; k0 += KB) {
    // A tile: rows m0..m0+127, cols k0..k0+15 (8 contiguous f32 per thread)
    {
      const float* src = Ai + (size_t)(m0 + arow) * NDIM + (k0 + acol);
#pragma unroll
      for (int j = 0; j < 8; ++j) As[arow][acol + j] = src[j];
    }
    if (!transB) {
      // Bs[r][c] = B[k0+r][n0+c]
      const float* src = Bi + (size_t)(k0 + brow) * NDIM + (n0 + bcol);
#pragma unroll
      for (int j = 0; j < 4; ++j) Bs[brow][bcol + j] = src[j];
    } else {
      // Bs[r][c] = B[(n0+c)*NDIM + k0+r]   (B holds D; effective D^T)
      const float* src = Bi + (size_t)(n0 + tcol) * NDIM + (k0 + trow);
#pragma unroll
      for (int j = 0; j < 4; ++j) Bs[trow + j][tcol] = src[j];
    }
    __syncthreads();

#pragma unroll
    for (int kk = 0; kk < KB; kk += 4) {
      v2f a[2], b[2];
      // A 16x4 fragment: lanes 0-15 -> K=kk,kk+1 ; lanes 16-31 -> K=kk+2,kk+3
#pragma unroll
      for (int mi = 0; mi < 2; ++mi) {
        int r = wm * 32 + mi * 16 + lr;
        a[mi].x = As[r][kk + 2 * hi];
        a[mi].y = As[r][kk + 2 * hi + 1];
      }
      // B 4x16 fragment: row-striped across lanes within VGPR
#pragma unroll
      for (int ni = 0; ni < 2; ++ni) {
        int c = wn * 32 + ni * 16 + lr;
        b[ni].x = Bs[kk + 2 * hi][c];
        b[ni].y = Bs[kk + 2 * hi + 1][c];
      }
#pragma unroll
      for (int mi = 0; mi < 2; ++mi)
#pragma unroll
        for (int ni = 0; ni < 2; ++ni)
          acc[mi][ni] = __builtin_amdgcn_wmma_f32_16x16x4_f32(
              false, a[mi], false, b[ni], (short)0, acc[mi][ni], false, false);
    }
    __syncthreads();
  }

  // epilogue: C/D layout -> VGPR e, lanes 0-15: M=e ; lanes 16-31: M=8+e
  float lmax = 0.0f;
#pragma unroll
  for (int mi = 0; mi < 2; ++mi) {
#pragma unroll
    for (int ni = 0; ni < 2; ++ni) {
      int row0 = m0 + wm * 32 + mi * 16 + 8 * hi;
      int col = n0 + wn * 32 + ni * 16 + lr;
#pragma unroll
      for (int e = 0; e < 8; ++e) {
        float v = acc[mi][ni][e];
        if (absMax) {
          v = fabsf(v);
          lmax = fmaxf(lmax, v);
        }
        Ci[(size_t)(row0 + e) * NDIM + col] = v;
      }
    }
  }

  if (absMax) {
    red[tid] = lmax;
    __syncthreads();
    for (int s = 128; s > 0; s >>= 1) {
      if (tid < s) red[tid] = fmaxf(red[tid], red[tid + s]);
      __syncthreads();
    }
    // mags are >= 0, so int-bit-pattern compare == float compare
    if (tid == 0) atomicMax(maxv + img, __float_as_int(red[0]));
  }
}

// ---------------------------------------------------------------------------
// K3: per-image 256-bin histogram (LDS ds_add atomics + global int atomics)
// grid = (64 chunks, 96 images); 256 threads x 16 elems = 4096 per block
// ---------------------------------------------------------------------------
__global__ __launch_bounds__(256) void dct_hist_kernel(
    const float* __restrict__ mag, const int* __restrict__ maxv,
    int* __restrict__ hist) {
  __shared__ int bins[NUM_BINS];
  const int img = blockIdx.y;
  const int tid = threadIdx.x;
  if (tid < NUM_BINS) bins[tid] = 0;
  __syncthreads();

  float m = fmaxf(__int_as_float(maxv[img]), 1e-12f);
  const float* p =
      mag + (size_t)img * NDIM * NDIM + (size_t)blockIdx.x * 4096 + tid;
#pragma unroll
  for (int j = 0; j < 16; ++j) {
    float v = p[j * 256];
    int idx = (int)floorf((v / m) * (float)NUM_BINS);
    idx = min(max(idx, 0), NUM_BINS - 1);
    atomicAdd(&bins[idx], 1);
  }
  __syncthreads();
  if (tid < NUM_BINS)
    atomicAdd(&hist[img * NUM_BINS + tid], bins[tid]);
}

// ---------------------------------------------------------------------------
// K4: fc1 -> LayerNorm -> ReLU -> fc2, one block per sample (96 blocks)
// ---------------------------------------------------------------------------
__global__ __launch_bounds__(256) void dct_mlp_kernel(
    const int* __restrict__ hist, const float* __restrict__ w1,
    const float* __restrict__ b1, const float* __restrict__ gamma,
    const float* __restrict__ beta, const float* __restrict__ w2,
    const float* __restrict__ b2, float* __restrict__ out) {
  __shared__ float sh_hist[NUM_BINS];
  __shared__ float sh_h[HIDDEN];
  __shared__ float red[256];
  const int n = blockIdx.x;
  const int tid = threadIdx.x;
  const float invHW = 1.0f / (float)(NDIM * NDIM);

  if (tid < NUM_BINS)
    sh_hist[tid] = (float)hist[n * NUM_BINS + tid] * invHW;
  __syncthreads();

  // fc1: two hidden units per thread
#pragma unroll
  for (int jj = 0; jj < 2; ++jj) {
    int j = tid + jj * 256;
    const float* wr = w1 + (size_t)j * NUM_BINS;
    float s = b1[j];
    for (int i = 0; i < NUM_BINS; ++i) s += wr[i] * sh_hist[i];
    sh_h[j] = s;
  }
  __syncthreads();

  // mean
  red[tid] = sh_h[tid] + sh_h[tid + 256];
  __syncthreads();
  for (int s = 128; s > 0; s >>= 1) {
    if (tid < s) red[tid] += red[tid + s];
    __syncthreads();
  }
  float mu = red[0] / (float)HIDDEN;
  __syncthreads();

  // variance (two-pass, matches reference form)
  float d0 = sh_h[tid] - mu, d1 = sh_h[tid + 256] - mu;
  red[tid] = d0 * d0 + d1 * d1;
  __syncthreads();
  for (int s = 128; s > 0; s >>= 1) {
    if (tid < s) red[tid] += red[tid + s];
    __syncthreads();
  }
  float rstd = 1.0f / sqrtf(red[0] / (float)HIDDEN + 1e-5f);
  __syncthreads();

  // LN + ReLU
#pragma unroll
  for (int jj = 0; jj < 2; ++jj) {
    int j = tid + jj * 256;
    float hn = (sh_h[j] - mu) * rstd * gamma[j] + beta[j];
    sh_h[j] = fmaxf(hn, 0.0f);
  }
  __syncthreads();

  // fc2: one output per thread
  {
    const float* wr = w2 + (size_t)tid * HIDDEN;
    float s = b2[tid];
    for (int j = 0; j < HIDDEN; ++j) s += wr[j] * sh_h[j];
    out[(size_t)n * FEATURE_DIM + tid] = s;
  }
}

// ---------------------------------------------------------------------------
extern "C" void kernel_launch(void* const* d_in, const int* in_sizes, int n_in,
                              void* d_out, int out_size, void* d_ws,
                              size_t ws_size, hipStream_t stream) {
  const float* x = (const float*)d_in[0];      // [32,3,512,512]
  const float* w1 = (const float*)d_in[1];     // [512,256]
  const float* b1 = (const float*)d_in[2];     // [512]
  const float* gamma = (const float*)d_in[3];  // [512]
  const float* beta = (const float*)d_in[4];   // [512]
  const float* w2 = (const float*)d_in[5];     // [256,512]
  const float* b2 = (const float*)d_in[6];     // [256]
  float* out = (float*)d_out;                  // [96,256]

  char* ws = (char*)d_ws;
  size_t off = 0;
  float* D = (float*)(ws + off);    off += (size_t)NDIM * NDIM * 4;        // 1 MB
  float* tmp = (float*)(ws + off);  off += (size_t)BC * NDIM * NDIM * 4;   // 100 MB
  float* mag = (float*)(ws + off);  off += (size_t)BC * NDIM * NDIM * 4;   // 100 MB
  int* maxv = (int*)(ws + off);     off += 512;
  int* hist = (int*)(ws + off);     off += (size_t)BC * NUM_BINS * 4;
  (void)ws_size; (void)in_sizes; (void)n_in; (void)out_size;

  // K0: DCT matrix + zero max/hist
  dct_init_kernel<<<dim3((NDIM * NDIM) / 256), 256, 0, stream>>>(D, maxv, hist);

  dim3 gemmGrid(NDIM / TILE_N, NDIM / TILE_M, BC);
  // K1: tmp[n] = D @ x[n]
  dct_gemm_kernel<<<gemmGrid, 256, 0, stream>>>(
      D, x, tmp, 0L, (long)NDIM * NDIM, (long)NDIM * NDIM, 0, 0, nullptr);
  // K2: mag[n] = |tmp[n] @ D^T|, maxv[n] = max(mag[n])
  dct_gemm_kernel<<<gemmGrid, 256, 0, stream>>>(
      tmp, D, mag, (long)NDIM * NDIM, 0L, (long)NDIM * NDIM, 1, 1, maxv);

  // K3: histograms
  dct_hist_kernel<<<dim3(64, BC), 256, 0, stream>>>(mag, maxv, hist);

  // K4: MLP head
  dct_mlp_kernel<<<BC, 256, 0, stream>>>(hist, w1, b1, gamma, beta, w2, b2,
                                         out);
}